// Hyperbolic_Lines_77730318123528
// MI455X (gfx1250) — compile-verified
//
#include <hip/hip_runtime.h>

typedef __attribute__((ext_vector_type(2))) float v2f;
typedef __attribute__((ext_vector_type(8))) float v8f;

#define DIMS 256
#define BLOCK 256
#define WAVES_PER_BLOCK (BLOCK / 32)
#define MAX_BLOCKS 1024

// ---------------- Kernel 1: norm_sq(w) -> ws[0] ----------------
__global__ __launch_bounds__(BLOCK) void hl_prep(const float* __restrict__ w,
                                                 float* __restrict__ ws) {
  __shared__ float red[BLOCK];
  const int t = threadIdx.x;
  float v = (t < DIMS) ? w[t] : 0.0f;
  red[t] = v * v;
  __syncthreads();
  for (int s = BLOCK / 2; s > 0; s >>= 1) {
    if (t < s) red[t] += red[t + s];
    __syncthreads();
  }
  if (t == 0) ws[0] = red[0];
}

// ---------------- Kernel 2: fused streaming + WMMA GEMV ----------------
// Each wave owns 16-row tiles. WMMA F32 16x16x4 accumulates s_m = y_m . w
// (B = w replicated across 16 columns, so every column of D equals s_m).
// q_m = ||y_m||^2 accumulated with VALU FMAs on the same loaded data.
__global__ __launch_bounds__(BLOCK) void hl_main(const float* __restrict__ w,
                                                 const float* __restrict__ y,
                                                 const float* __restrict__ ws,
                                                 float* __restrict__ block_sums,
                                                 int n) {
  __shared__ float s_w[DIMS];                       // 1 KB broadcast copy of w
  __shared__ float s_tile[WAVES_PER_BLOCK][16 * 16]; // 1 KB per wave: D extraction
  __shared__ float s_red[WAVES_PER_BLOCK];

  const int t = threadIdx.x;
  const int lane = t & 31;
  const int wib = t >> 5;

  for (int i = t; i < DIMS; i += BLOCK) s_w[i] = w[i];
  __syncthreads();

  const float inv_nsq = 1.0f / ws[0];

  const int rowInTile = lane & 15;          // M index this lane carries for A
  const int kgrp = (lane >> 4) << 1;        // K offset within each K=4 step: 0 or 2

  const int waveId = blockIdx.x * WAVES_PER_BLOCK + wib;
  const int numWaves = gridDim.x * WAVES_PER_BLOCK;
  const int numTiles = (n + 15) >> 4;

  float acc = 0.0f;
  float* myTile = &s_tile[wib][0];

  for (int tile = waveId; tile < numTiles; tile += numWaves) {
    const int row0 = tile << 4;
    int r = row0 + rowInTile;
    if (r >= n) r = n - 1;                  // clamp (masked out later); n%16==0 anyway
    const float* rowp = y + (size_t)r * DIMS + kgrp;

    // Prefetch this lane's row for the next tile (global_prefetch_b8).
    const int nextTile = tile + numWaves;
    if (nextTile < numTiles) {
      const int rn = (nextTile << 4) + rowInTile;
      __builtin_prefetch(y + (size_t)rn * DIMS + kgrp, 0, 1);
    }

    v8f c0 = {0, 0, 0, 0, 0, 0, 0, 0};
    v8f c1 = {0, 0, 0, 0, 0, 0, 0, 0};
    float qp = 0.0f;

#pragma unroll 4
    for (int kb = 0; kb < DIMS; kb += 8) {
      // A chunk: lane L holds Y[row0 + L%16, kb + kgrp + {0,1}]  (ISA 16x4 f32 layout)
      v2f a0 = *(const v2f*)(rowp + kb);
      v2f a1 = *(const v2f*)(rowp + kb + 4);
      // B chunk: w[k] replicated across all 16 columns (same K mapping as A)
      v2f b0; b0.x = s_w[kb + kgrp];     b0.y = s_w[kb + kgrp + 1];
      v2f b1; b1.x = s_w[kb + 4 + kgrp]; b1.y = s_w[kb + 4 + kgrp + 1];

      qp = fmaf(a0.x, a0.x, qp);
      qp = fmaf(a0.y, a0.y, qp);
      qp = fmaf(a1.x, a1.x, qp);
      qp = fmaf(a1.y, a1.y, qp);

      c0 = __builtin_amdgcn_wmma_f32_16x16x4_f32(false, a0, false, b0,
                                                 (short)0, c0, false, false);
      c1 = __builtin_amdgcn_wmma_f32_16x16x4_f32(false, a1, false, b1,
                                                 (short)0, c1, false, false);
    }
    v8f c = c0 + c1;

    // Lane L covered K = {kgrp, kgrp+1} mod 4 of row L%16; partner lane covers rest.
    qp += __shfl_xor(qp, 16);             // lanes 0..15 now hold full q for row (lane&15)

    // Extract s_m: spill D tile (every column equals s_m) to wave-private LDS,
    // then lane m reads element [m][m]. Same-wave LDS ops are in-order.
    const int Mbase = (lane < 16) ? 0 : 8;
    const int N = lane & 15;
#pragma unroll
    for (int v = 0; v < 8; ++v)
      myTile[(Mbase + v) * 16 + N] = c[v];

    const float s = myTile[rowInTile * 16 + rowInTile];

    if (lane < 16) {
      const int rr = row0 + rowInTile;
      if (rr < n) {
        const float dist = fmaf(-(s * inv_nsq), s, qp);   // q - s^2/||w||^2
        const float z = 1.0f + dist;
        float t2 = fmaf(z, z, -1.0f);
        t2 = (t2 > 0.0f) ? t2 : 0.0f;
        const float ach = logf(z + sqrtf(t2));            // acosh as in reference
        acc = fmaf(ach, ach, acc);
      }
    }
  }

  // Wave reduction, then per-block partial sum (deterministic final pass later).
#pragma unroll
  for (int off = 16; off > 0; off >>= 1) acc += __shfl_down(acc, off);
  if (lane == 0) s_red[wib] = acc;
  __syncthreads();
  if (t == 0) {
    float tot = 0.0f;
#pragma unroll
    for (int i = 0; i < WAVES_PER_BLOCK; ++i) tot += s_red[i];
    block_sums[blockIdx.x] = tot;
  }
}

// ---------------- Kernel 3: deterministic final reduction ----------------
__global__ __launch_bounds__(BLOCK) void hl_final(const float* __restrict__ block_sums,
                                                  int nblocks,
                                                  float* __restrict__ out) {
  __shared__ float red[BLOCK];
  const int t = threadIdx.x;
  float v = 0.0f;
  for (int i = t; i < nblocks; i += BLOCK) v += block_sums[i];
  red[t] = v;
  __syncthreads();
  for (int s = BLOCK / 2; s > 0; s >>= 1) {
    if (t < s) red[t] += red[t + s];
    __syncthreads();
  }
  if (t == 0) out[0] = red[0];
}

extern "C" void kernel_launch(void* const* d_in, const int* in_sizes, int n_in,
                              void* d_out, int out_size, void* d_ws, size_t ws_size,
                              hipStream_t stream) {
  const float* w = (const float*)d_in[0];
  const float* y = (const float*)d_in[1];
  float* out = (float*)d_out;
  float* ws = (float*)d_ws;          // ws[0] = ||w||^2 ; ws[1..] = block partial sums
  const int n = in_sizes[1] / DIMS;

  const int numTiles = (n + 15) >> 4;
  int blocks = (numTiles + WAVES_PER_BLOCK - 1) / WAVES_PER_BLOCK;
  if (blocks > MAX_BLOCKS) blocks = MAX_BLOCKS;
  if (blocks < 1) blocks = 1;

  hl_prep<<<1, BLOCK, 0, stream>>>(w, ws);
  hl_main<<<blocks, BLOCK, 0, stream>>>(w, y, ws, ws + 1, n);
  hl_final<<<1, BLOCK, 0, stream>>>(ws + 1, blocks, out);
}